// TESTMODEL_74998718923374
// MI455X (gfx1250) — compile-verified
//
#include <hip/hip_runtime.h>
#include <cmath>

typedef __attribute__((ext_vector_type(2))) float v2f;
typedef __attribute__((ext_vector_type(8))) float v8f;

#define HDIM 128   // all GEMM outputs are 128 columns wide

// ---------------------------------------------------------------------------
// Y[M x 128] = act( X [M x K] @ W [K x 128] (+ bias) )
// One wave computes one 16x16 output tile via V_WMMA_F32_16X16X4_F32 (exact
// fp32). Block = (32,8): 8 waves cover a 16-row x 128-col output strip.
// A/B loads use hoisted per-lane base pointers + constant offsets so the
// unrolled inner loop is load/WMMA only (no per-step 64-bit address math).
// ---------------------------------------------------------------------------
__global__ __launch_bounds__(256) void gemm_wmma_f32(
    const float* __restrict__ X, const float* __restrict__ W,
    const float* __restrict__ bias, float* __restrict__ Y,
    int M, int K, int relu)
{
    const int lane = threadIdx.x;         // 0..31
    const int hi   = lane >> 4;           // 0 / 1
    const int m    = lane & 15;
    const int row0 = blockIdx.x << 4;
    const int col  = (threadIdx.y << 4) + m;   // 0..127

    int arowi = row0 + m;
    if (arowi >= M) arowi = M - 1;        // clamp (no EXEC divergence)

    // per-lane base pointers; all k-loop offsets are compile-time affine
    const float* ap = X + (size_t)arowi * (size_t)K + (hi << 1);
    const float* wp = W + (size_t)(hi << 1) * HDIM + col;

    v8f acc = {};
    const int steps = K >> 2;             // K=4 per WMMA
#pragma unroll 8
    for (int kk = 0; kk < steps; ++kk) {
        v2f a = *(const v2f*)(ap + (kk << 2));       // 8B load, const offset
        v2f b;
        b.x = wp[(kk << 9)];                          // kk*4*HDIM
        b.y = wp[(kk << 9) + HDIM];
        acc = __builtin_amdgcn_wmma_f32_16x16x4_f32(
            false, a, false, b, (short)0, acc, false, false);
    }

    const float bv = bias ? bias[col] : 0.0f;
#pragma unroll
    for (int v = 0; v < 8; ++v) {
        const int row = row0 + v + (hi << 3);
        if (row < M) {
            float val = acc[v] + bv;
            if (relu) val = fmaxf(val, 0.0f);
            Y[(size_t)row * HDIM + col] = val;
        }
    }
}

// ---------------------------------------------------------------------------
// CSR-by-destination construction (count -> scan -> place)
// ---------------------------------------------------------------------------
__global__ void fill_i32(int* __restrict__ p, int n, int v) {
    int i = blockIdx.x * 256 + threadIdx.x;
    if (i < n) p[i] = v;
}

// pos[e] = slot of edge e within its destination's list; cnt[d] = in-degree
__global__ void csr_count(const long long* __restrict__ dst,
                          int* __restrict__ cnt, int* __restrict__ pos, int E) {
    int e = blockIdx.x * 256 + threadIdx.x;
    if (e < E) pos[e] = atomicAdd(&cnt[(int)dst[e]], 1);
}

// per-256-block sums of cnt
__global__ void scan_partial(const int* __restrict__ cnt, int* __restrict__ bsum, int N) {
    __shared__ int sh[256];
    int i = blockIdx.x * 256 + threadIdx.x;
    sh[threadIdx.x] = (i < N) ? cnt[i] : 0;
    __syncthreads();
    for (int off = 128; off > 0; off >>= 1) {
        if (threadIdx.x < off) sh[threadIdx.x] += sh[threadIdx.x + off];
        __syncthreads();
    }
    if (threadIdx.x == 0) bsum[blockIdx.x] = sh[0];
}

// serial exclusive scan of the (≈200) block sums
__global__ void scan_bsums(int* __restrict__ bsum, int nb) {
    if (blockIdx.x == 0 && threadIdx.x == 0) {
        int run = 0;
        for (int b = 0; b < nb; ++b) { int t = bsum[b]; bsum[b] = run; run += t; }
    }
}

// rowstart[i] = exclusive prefix over cnt ; rowstart[N] = E
__global__ void scan_final(const int* __restrict__ cnt, const int* __restrict__ boff,
                           int* __restrict__ rowstart, int N, int E) {
    __shared__ int sh[256];
    int i = blockIdx.x * 256 + threadIdx.x;
    int v = (i < N) ? cnt[i] : 0;
    sh[threadIdx.x] = v;
    __syncthreads();
    for (int off = 1; off < 256; off <<= 1) {
        int t = (threadIdx.x >= off) ? sh[threadIdx.x - off] : 0;
        __syncthreads();
        sh[threadIdx.x] += t;
        __syncthreads();
    }
    if (i < N) rowstart[i] = boff[blockIdx.x] + sh[threadIdx.x] - v;
    if (i == N - 1) rowstart[N] = E;
}

__global__ void csr_fill(const long long* __restrict__ src,
                         const long long* __restrict__ dst,
                         const int* __restrict__ rowstart,
                         const int* __restrict__ pos,
                         int* __restrict__ colIdx, int E) {
    int e = blockIdx.x * 256 + threadIdx.x;
    if (e < E) colIdx[rowstart[(int)dst[e]] + pos[e]] = (int)src[e];
}

// dinv = rsqrt(indegree + 1)  (self loop)
__global__ void make_dinv(const int* __restrict__ cnt, float* __restrict__ dinv, int N) {
    int i = blockIdx.x * 256 + threadIdx.x;
    if (i < N) dinv[i] = rsqrtf((float)cnt[i] + 1.0f);
}

// ---------------------------------------------------------------------------
// GIN gather: out[i] = feat[i] + sum_{s in Nin(i)} feat[s]   (no atomics)
// one wave per node, float4 per lane; gathered matrix is L2-resident (25.6MB).
// Index load is software-pipelined one edge ahead of the feature-row load.
// ---------------------------------------------------------------------------
__global__ __launch_bounds__(256) void gin_gather(
    const float* __restrict__ feat, float* __restrict__ out,
    const int* __restrict__ rowstart, const int* __restrict__ colIdx, int N)
{
    int i = blockIdx.x * blockDim.y + threadIdx.y;
    if (i >= N) return;
    const int c  = threadIdx.x * 4;
    const int s0 = rowstart[i], s1 = rowstart[i + 1];
    float4 acc = *(const float4*)(feat + (size_t)i * HDIM + c);   // self term
    if (s0 < s1) {
        int s = colIdx[s0];
        for (int j = s0; j + 1 < s1; ++j) {
            const int sn = colIdx[j + 1];                 // in flight with row
            const float4 v = *(const float4*)(feat + (size_t)s * HDIM + c);
            acc.x += v.x; acc.y += v.y; acc.z += v.z; acc.w += v.w;
            s = sn;
        }
        const float4 v = *(const float4*)(feat + (size_t)s * HDIM + c);
        acc.x += v.x; acc.y += v.y; acc.z += v.z; acc.w += v.w;
    }
    *(float4*)(out + (size_t)i * HDIM + c) = acc;
}

// ---------------------------------------------------------------------------
// fused GCN gather + finish:
// out[i] = relu( dinv[i]*sum_{s} dinv[s]*xw[s] + dinv[i]^2*xw[i] + b )
// ---------------------------------------------------------------------------
__global__ __launch_bounds__(256) void gcn_gather(
    const float* __restrict__ xw, float* __restrict__ out,
    const int* __restrict__ rowstart, const int* __restrict__ colIdx,
    const float* __restrict__ dinv, const float* __restrict__ bias, int N)
{
    int i = blockIdx.x * blockDim.y + threadIdx.y;
    if (i >= N) return;
    const int c  = threadIdx.x * 4;
    const int s0 = rowstart[i], s1 = rowstart[i + 1];
    float4 acc = make_float4(0.f, 0.f, 0.f, 0.f);
    if (s0 < s1) {
        int s = colIdx[s0];
        for (int j = s0; j + 1 < s1; ++j) {
            const int sn = colIdx[j + 1];
            const float ds = dinv[s];
            const float4 v = *(const float4*)(xw + (size_t)s * HDIM + c);
            acc.x += v.x * ds; acc.y += v.y * ds; acc.z += v.z * ds; acc.w += v.w * ds;
            s = sn;
        }
        const float ds = dinv[s];
        const float4 v = *(const float4*)(xw + (size_t)s * HDIM + c);
        acc.x += v.x * ds; acc.y += v.y * ds; acc.z += v.z * ds; acc.w += v.w * ds;
    }
    const float  di  = dinv[i];
    const float  dii = di * di;
    const float4 self = *(const float4*)(xw + (size_t)i * HDIM + c);
    float4 o;
    o.x = fmaxf(di * acc.x + dii * self.x + bias[c + 0], 0.f);
    o.y = fmaxf(di * acc.y + dii * self.y + bias[c + 1], 0.f);
    o.z = fmaxf(di * acc.z + dii * self.z + bias[c + 2], 0.f);
    o.w = fmaxf(di * acc.w + dii * self.w + bias[c + 3], 0.f);
    *(float4*)(out + (size_t)i * HDIM + c) = o;
}

// ---------------------------------------------------------------------------
// mean pool: batch is sorted, so graph g is a contiguous node range.
// one block (128 threads) per graph; binary search the segment; no atomics.
// ---------------------------------------------------------------------------
__global__ __launch_bounds__(128) void pool_seg(
    const float* __restrict__ xfeat, float* __restrict__ pooled,
    const long long* __restrict__ batch, int colOff, int N)
{
    const int g = blockIdx.x;
    int lo = 0, hi = N;
    while (lo < hi) { int mid = (lo + hi) >> 1; if (batch[mid] < (long long)g) lo = mid + 1; else hi = mid; }
    const int start = lo;
    hi = N;
    while (lo < hi) { int mid = (lo + hi) >> 1; if (batch[mid] < (long long)(g + 1)) lo = mid + 1; else hi = mid; }
    const int end = lo;

    const int c = threadIdx.x;            // 0..127
    float s = 0.f;
    for (int n = start; n < end; ++n) s += xfeat[(size_t)n * HDIM + c];
    pooled[(size_t)g * 384 + colOff + c] = s / fmaxf((float)(end - start), 1.0f);
}

// ---------------------------------------------------------------------------
// row-wise L2 normalize: one wave per row, wave32 shuffle reduction
// ---------------------------------------------------------------------------
__global__ __launch_bounds__(256) void l2norm_rows(
    const float* __restrict__ Z, float* __restrict__ out, int rows)
{
    int r = blockIdx.x * blockDim.y + threadIdx.y;
    if (r >= rows) return;
    const int c = threadIdx.x * 4;
    float4 v = *(const float4*)(Z + (size_t)r * HDIM + c);
    float s = v.x * v.x + v.y * v.y + v.z * v.z + v.w * v.w;
    for (int off = 16; off > 0; off >>= 1) s += __shfl_xor(s, off, 32);
    const float inv = 1.0f / fmaxf(sqrtf(s), 1e-12f);
    float4 o; o.x = v.x * inv; o.y = v.y * inv; o.z = v.z * inv; o.w = v.w * inv;
    *(float4*)(out + (size_t)r * HDIM + c) = o;
}

// ---------------------------------------------------------------------------
extern "C" void kernel_launch(void* const* d_in, const int* in_sizes, int n_in,
                              void* d_out, int out_size, void* d_ws, size_t ws_size,
                              hipStream_t stream)
{
    const float*     x     = (const float*)d_in[0];
    const long long* ei    = (const long long*)d_in[1];   // int64 [2,E]
    const long long* batch = (const long long*)d_in[2];   // int64 [N], sorted
    const float *W_in = (const float*)d_in[3],  *b_in = (const float*)d_in[4];
    const float *W_g1 = (const float*)d_in[5],  *b_g1 = (const float*)d_in[6];
    const float *W_g2 = (const float*)d_in[7],  *b_g2 = (const float*)d_in[8];
    const float *W_c1 = (const float*)d_in[9],  *b_c1 = (const float*)d_in[10];
    const float *W_c2 = (const float*)d_in[11], *b_c2 = (const float*)d_in[12];
    const float *W_p1 = (const float*)d_in[13], *b_p1 = (const float*)d_in[14];
    const float *W_p2 = (const float*)d_in[15], *b_p2 = (const float*)d_in[16];
    float* out = (float*)d_out;

    const int N = in_sizes[0] / HDIM;   // 50000
    const int E = in_sizes[1] / 2;      // 600000
    const int G = out_size / HDIM;      // 512
    const long long* srcI = ei;
    const long long* dstI = ei + E;

    // ---- workspace layout (bytes, 256-aligned chunks) ----
    char* base = (char*)d_ws;
    size_t off = 0;
    auto take = [&](size_t bytes) { void* p = base + off; off += (bytes + 255) & ~(size_t)255; return p; };
    const size_t NB = (size_t)N * HDIM;
    float* bufA    = (float*)take(NB * 4);            // h -> xw (each layer)
    float* bufB    = (float*)take(NB * 4);            // gin-sum -> x1 -> x2 -> x3
    float* dinv    = (float*)take((size_t)N * 4);
    float* pooled  = (float*)take((size_t)G * 384 * 4);
    float* p1      = (float*)take((size_t)G * HDIM * 4);
    float* z       = (float*)take((size_t)G * HDIM * 4);
    int*   cnt     = (int*)take((size_t)N * 4);
    int*   rowstart= (int*)take((size_t)(N + 1) * 4);
    int*   pos     = (int*)take((size_t)E * 4);
    int*   colIdx  = (int*)take((size_t)E * 4);
    const int nb   = (N + 255) / 256;
    int*   bsum    = (int*)take((size_t)nb * 4);
    (void)ws_size; (void)n_in;

    const dim3 blk(32, 8);                  // 8 waves
    const int gemmN = (N + 15) / 16;
    const int gemmG = (G + 15) / 16;
    const int nodeB = (N + 7) / 8;
    const int eB    = (E + 255) / 256;

    // ---- CSR by destination (feature-independent; also yields degrees) ----
    fill_i32<<<nb, 256, 0, stream>>>(cnt, N, 0);
    csr_count<<<eB, 256, 0, stream>>>(dstI, cnt, pos, E);
    scan_partial<<<nb, 256, 0, stream>>>(cnt, bsum, N);
    scan_bsums<<<1, 1, 0, stream>>>(bsum, nb);
    scan_final<<<nb, 256, 0, stream>>>(cnt, bsum, rowstart, N, E);
    csr_fill<<<eB, 256, 0, stream>>>(srcI, dstI, rowstart, pos, colIdx, E);
    make_dinv<<<nb, 256, 0, stream>>>(cnt, dinv, N);

    // ---- 1) input projection: h = x @ W_in + b_in ----
    gemm_wmma_f32<<<gemmN, blk, 0, stream>>>(x, W_in, b_in, bufA, N, HDIM, 0);

    // ---- 2) GIN: s = h + sum_N(h); t = relu(s@W_g1+b); x1 = t@W_g2+b ----
    gin_gather<<<nodeB, blk, 0, stream>>>(bufA, bufB, rowstart, colIdx, N);
    gemm_wmma_f32<<<gemmN, blk, 0, stream>>>(bufB, W_g1, b_g1, bufA, N, HDIM, 1);
    gemm_wmma_f32<<<gemmN, blk, 0, stream>>>(bufA, W_g2, b_g2, bufB, N, HDIM, 0); // x1
    pool_seg<<<G, 128, 0, stream>>>(bufB, pooled, batch, 0, N);

    // ---- 3) GCN layer 1: x2 = relu(norm-agg(x1@W_c1) + b_c1) ----
    gemm_wmma_f32<<<gemmN, blk, 0, stream>>>(bufB, W_c1, nullptr, bufA, N, HDIM, 0);
    gcn_gather<<<nodeB, blk, 0, stream>>>(bufA, bufB, rowstart, colIdx, dinv, b_c1, N); // x2
    pool_seg<<<G, 128, 0, stream>>>(bufB, pooled, batch, HDIM, N);

    // ---- 4) GCN layer 2: x3 ----
    gemm_wmma_f32<<<gemmN, blk, 0, stream>>>(bufB, W_c2, nullptr, bufA, N, HDIM, 0);
    gcn_gather<<<nodeB, blk, 0, stream>>>(bufA, bufB, rowstart, colIdx, dinv, b_c2, N); // x3
    pool_seg<<<G, 128, 0, stream>>>(bufB, pooled, batch, 2 * HDIM, N);

    // ---- 5) projection head + normalize ----
    gemm_wmma_f32<<<gemmG, blk, 0, stream>>>(pooled, W_p1, b_p1, p1, G, 3 * HDIM, 1);
    gemm_wmma_f32<<<gemmG, blk, 0, stream>>>(p1, W_p2, b_p2, z, G, HDIM, 0);
    l2norm_rows<<<(G + 7) / 8, blk, 0, stream>>>(z, out, G);
}